// EdgeConvGat_supervised_35485019800325
// MI455X (gfx1250) — compile-verified
//
#include <hip/hip_runtime.h>
#include <hip/hip_bf16.h>

typedef __attribute__((ext_vector_type(2))) float v2f;
typedef __attribute__((ext_vector_type(4))) float v4f;
typedef __attribute__((ext_vector_type(8))) float v8f;

#define Bsz 8
#define Nn  256
#define Ff  128
#define Ee  32

// ---------------------------------------------------------------------------
// K1: W_combo[e,f] = sum_k W_edge[e,k]*W_emb[128+k,f]   (blocks 0..31)
//     const_vec[f] = b_emb[f] + sum_k b_edge[k]*W_emb[128+k,f]  (block 32)
// ---------------------------------------------------------------------------
__global__ void k_prep(const float* __restrict__ W_edge, const float* __restrict__ b_edge,
                       const float* __restrict__ W_emb, const float* __restrict__ b_emb,
                       float* __restrict__ W_combo, float* __restrict__ const_vec) {
    int f = threadIdx.x;            // 0..127
    int e = blockIdx.x;             // 0..32
    if (e < Ee) {
        float acc = 0.f;
        for (int k = 0; k < Ff; ++k)
            acc += W_edge[e * Ff + k] * W_emb[(Ff + k) * Ff + f];
        W_combo[e * Ff + f] = acc;
    } else {
        float acc = b_emb[f];
        for (int k = 0; k < Ff; ++k)
            acc += b_edge[k] * W_emb[(Ff + k) * Ff + f];
        const_vec[f] = acc;
    }
}

// ---------------------------------------------------------------------------
// K2: c[row,f] = const_vec[f] + sum_k x[row,k]*W_emb[k,f]   row = b*N+j
// ---------------------------------------------------------------------------
__global__ void k_cmat(const float* __restrict__ x, const float* __restrict__ W_emb,
                       const float* __restrict__ const_vec, float* __restrict__ cmat) {
    int row = blockIdx.x;           // 0..2047
    int f   = threadIdx.x;          // 0..127
    const float* xr = x + (size_t)row * Ff;
    float acc = const_vec[f];
    for (int k = 0; k < Ff; ++k)
        acc += xr[k] * W_emb[k * Ff + f];
    cmat[(size_t)row * Ff + f] = acc;
}

// ---------------------------------------------------------------------------
// K3: ea[row,e] = sum_j adj[row,j] * edge_attrs[row,j,e]   (HBM-bound stage)
// 256 threads/block, one block per (b,i) row. Each lane owns 4 consecutive
// E-channels -> global_load_b128 streaming, 4 KB contiguous per iteration.
// ---------------------------------------------------------------------------
__global__ void k_ea(const float* __restrict__ edge_attrs, const float* __restrict__ adj,
                     float* __restrict__ ea) {
    __shared__ v4f red[256];
    int row = blockIdx.x;                   // b*N + i
    int t   = threadIdx.x;
    int e4  = (t & 7) * 4;                  // channel group: e4..e4+3
    int jg  = t >> 3;                       // 0..31
    const float* ep = edge_attrs + (size_t)row * (Nn * Ee) + e4;
    const float* ar = adj + (size_t)row * Nn;
    v4f acc = {0.f, 0.f, 0.f, 0.f};
    for (int j = jg; j < Nn; j += 32) {
        v4f v = *(const v4f*)(ep + (size_t)j * Ee);   // b128, 16B aligned
        acc += v * ar[j];
    }
    red[t] = acc;
    __syncthreads();
    if (t < 8) {
        v4f s = {0.f, 0.f, 0.f, 0.f};
        #pragma unroll
        for (int g = 0; g < 32; ++g) s += red[g * 8 + t];
        *(v4f*)(ea + (size_t)row * Ee + t * 4) = s;   // b128 store
    }
}

// ---------------------------------------------------------------------------
// K4: nws_b = adj_b @ c_b + ea_b @ W_combo, via V_WMMA_F32_16X16X4_F32.
// One wave per 16x16 output tile. grid = (F/16, N/16, B), block = 32.
// ---------------------------------------------------------------------------
__global__ void __launch_bounds__(32) k_nws(const float* __restrict__ adj,
                                            const float* __restrict__ cmat,
                                            const float* __restrict__ ea,
                                            const float* __restrict__ W_combo,
                                            float* __restrict__ nws) {
    int lane = threadIdx.x;
    int g  = lane >> 4;        // K half
    int mn = lane & 15;        // m for A fragment, n for B/D fragments
    int b  = blockIdx.z;
    int i0 = blockIdx.y * 16;
    int f0 = blockIdx.x * 16;

    v8f acc = {0.f, 0.f, 0.f, 0.f, 0.f, 0.f, 0.f, 0.f};

    // --- adj_b (256x256) @ c_b (256x128) ---
    const float* arow = adj  + ((size_t)b * Nn + i0 + mn) * Nn;        // A row m
    const float* ccol = cmat + ((size_t)b * Nn) * Ff + f0 + mn;        // B col n
    for (int k0 = 0; k0 < Nn; k0 += 4) {
        int k = k0 + g * 2;
        v2f a = *(const v2f*)(arow + k);                  // A[m][k], A[m][k+1]
        v2f bb;
        bb.x = ccol[(size_t)k * Ff];                      // B[k][n]
        bb.y = ccol[(size_t)(k + 1) * Ff];                // B[k+1][n]
        acc = __builtin_amdgcn_wmma_f32_16x16x4_f32(false, a, false, bb,
                                                    (short)0, acc, false, false);
    }

    // --- ea_b (256x32) @ W_combo (32x128) ---
    const float* erow = ea + ((size_t)b * Nn + i0 + mn) * Ee;
    const float* wcol = W_combo + f0 + mn;
    for (int k0 = 0; k0 < Ee; k0 += 4) {
        int k = k0 + g * 2;
        v2f a = *(const v2f*)(erow + k);
        v2f bb;
        bb.x = wcol[(size_t)k * Ff];
        bb.y = wcol[(size_t)(k + 1) * Ff];
        acc = __builtin_amdgcn_wmma_f32_16x16x4_f32(false, a, false, bb,
                                                    (short)0, acc, false, false);
    }

    // D: VGPR r -> row i0 + g*8 + r, col f0 + mn
    float* drow = nws + ((size_t)b * Nn + i0 + g * 8) * Ff + f0 + mn;
    #pragma unroll
    for (int r = 0; r < 8; ++r)
        drow[(size_t)r * Ff] = acc[r];
}

// ---------------------------------------------------------------------------
// K5: out = x @ W_fin[:128] + nws @ W_fin[128:] + b_fin   (2048x256x128 GEMM)
// grid = (F/16, (B*N)/16), block = 32 (one wave per tile).
// ---------------------------------------------------------------------------
__global__ void __launch_bounds__(32) k_out(const float* __restrict__ x,
                                            const float* __restrict__ nws,
                                            const float* __restrict__ W_fin,
                                            const float* __restrict__ b_fin,
                                            float* __restrict__ outp) {
    int lane = threadIdx.x;
    int g  = lane >> 4;
    int mn = lane & 15;
    int m0 = blockIdx.y * 16;      // flattened row b*N+i
    int f0 = blockIdx.x * 16;

    float bias = b_fin[f0 + mn];   // per-column bias, broadcast over rows
    v8f acc = {bias, bias, bias, bias, bias, bias, bias, bias};

    const float* xrow = x   + (size_t)(m0 + mn) * Ff;
    const float* nrow = nws + (size_t)(m0 + mn) * Ff;
    const float* wcol = W_fin + f0 + mn;

    for (int k0 = 0; k0 < Ff; k0 += 4) {           // K = 0..127 -> x half
        int k = k0 + g * 2;
        v2f a = *(const v2f*)(xrow + k);
        v2f bb;
        bb.x = wcol[(size_t)k * Ff];
        bb.y = wcol[(size_t)(k + 1) * Ff];
        acc = __builtin_amdgcn_wmma_f32_16x16x4_f32(false, a, false, bb,
                                                    (short)0, acc, false, false);
    }
    const float* wcol2 = wcol + Ff * Ff;           // K = 128..255 -> nws half
    for (int k0 = 0; k0 < Ff; k0 += 4) {
        int k = k0 + g * 2;
        v2f a = *(const v2f*)(nrow + k);
        v2f bb;
        bb.x = wcol2[(size_t)k * Ff];
        bb.y = wcol2[(size_t)(k + 1) * Ff];
        acc = __builtin_amdgcn_wmma_f32_16x16x4_f32(false, a, false, bb,
                                                    (short)0, acc, false, false);
    }

    float* drow = outp + (size_t)(m0 + g * 8) * Ff + f0 + mn;
    #pragma unroll
    for (int r = 0; r < 8; ++r)
        drow[(size_t)r * Ff] = acc[r];
}

// ---------------------------------------------------------------------------
extern "C" void kernel_launch(void* const* d_in, const int* in_sizes, int n_in,
                              void* d_out, int out_size, void* d_ws, size_t ws_size,
                              hipStream_t stream) {
    const float* x          = (const float*)d_in[0];
    const float* edge_attrs = (const float*)d_in[1];
    const float* adj        = (const float*)d_in[2];
    const float* W_edge     = (const float*)d_in[3];
    const float* b_edge     = (const float*)d_in[4];
    const float* W_emb      = (const float*)d_in[5];
    const float* b_emb      = (const float*)d_in[6];
    const float* W_fin      = (const float*)d_in[7];
    const float* b_fin      = (const float*)d_in[8];

    float* outp = (float*)d_out;                 // (B,N,F) = 262144
    float* nws  = outp + Bsz * Nn * Ff;          // second tuple output, also K5 input

    float* ws        = (float*)d_ws;
    float* W_combo   = ws;                       // 32*128    = 4096 floats
    float* const_vec = ws + 4096;                // 128 (padded to 4096)
    float* cmat      = ws + 8192;                // 2048*128  = 262144 floats
    float* eabuf     = ws + 8192 + Bsz * Nn * Ff;// 2048*32   = 65536 floats

    k_prep<<<Ee + 1, Ff, 0, stream>>>(W_edge, b_edge, W_emb, b_emb, W_combo, const_vec);
    k_cmat<<<Bsz * Nn, Ff, 0, stream>>>(x, W_emb, const_vec, cmat);
    k_ea  <<<Bsz * Nn, 256, 0, stream>>>(edge_attrs, adj, eabuf);
    k_nws <<<dim3(Ff / 16, Nn / 16, Bsz), 32, 0, stream>>>(adj, cmat, eabuf, W_combo, nws);
    k_out <<<dim3(Ff / 16, (Bsz * Nn) / 16), 32, 0, stream>>>(x, nws, W_fin, b_fin, outp);
}